// Net_53386443489635
// MI455X (gfx1250) — compile-verified
//
#include <hip/hip_runtime.h>
#include <hip/hip_bf16.h>
#include <math.h>

// ---------------------------------------------------------------------------
// SpiralNet++-style network on gfx1250 (MI455X), fp32 end-to-end.
// Spiral-conv / fc layers run on the matrix pipe via V_WMMA_F32_16X16X4_F32.
// Intermediates live in d_ws:
//   h0 [N,16] | h1 [N,32] | h2 [N,64] | h3 [N,128] | h4 [N,256]  (~198.4 MB)
// ---------------------------------------------------------------------------

typedef __attribute__((ext_vector_type(2))) float v2f;
typedef __attribute__((ext_vector_type(8))) float v8f;

// Branchless ELU: native v_exp_f32 (via __expf) + v_cndmask.  No EXEC-mask
// manipulation, ~4 VALU ops instead of libm expf's branchy range reduction.
__device__ __forceinline__ float elu(float v) {
    float e = __expf(v) - 1.0f;
    return v > 0.0f ? v : e;
}

// ---------------------------------------------------------------------------
// fc0: [N,3] -> [N,16] + ELU.  Tiny; plain VALU, one thread per node.
// ---------------------------------------------------------------------------
__global__ __launch_bounds__(256) void fc0_elu_kernel(
    const float* __restrict__ x, const float* __restrict__ w,
    const float* __restrict__ b, float* __restrict__ out, int n) {
    int i = blockIdx.x * blockDim.x + threadIdx.x;
    if (i >= n) return;
    float x0 = x[i * 3 + 0], x1 = x[i * 3 + 1], x2 = x[i * 3 + 2];
#pragma unroll
    for (int o = 0; o < 16; ++o) {
        float v = fmaf(w[o * 3 + 0], x0,
                  fmaf(w[o * 3 + 1], x1,
                  fmaf(w[o * 3 + 2], x2, b[o])));
        out[(size_t)i * 16 + o] = elu(v);
    }
}

// ---------------------------------------------------------------------------
// Spiral conv (and fc1 with S=1): out[n,o] = ELU( sum_k g[n,k]*w[o,k] + b[o] )
// where g[n, s*C_IN+c] = in[idx[n,s], c]   (HAS_IDX) or in[n,c] (S==1).
//
// One wave per 16-node tile; CHUNKS 16-wide output chunks accumulated in
// registers; K swept in steps of 4 with V_WMMA_F32_16X16X4_F32.
//
// f32 WMMA lane layouts (ISA 7.12.2), hi = lane>>4, m/n = lane&15:
//   A 16x4 : v0=A[m][2*hi], v1=A[m][2*hi+1]          -> one b64 gather
//   B 4x16 : v0=B[2*hi][n], v1=B[2*hi+1][n]          -> one b64 from W row
//   C 16x16: elem r of v8f  = (row r+8*hi, col n)    -> coalesced row stores
//
// K-loop unrolled x4 so constant byte offsets fold into the loads' IOFFSET
// field (amortizes 64-bit pointer bumps, more loads in flight against L2).
// ---------------------------------------------------------------------------
template <int C_IN, int C_OUT, int S, bool HAS_IDX, int CHUNKS>
__global__ __launch_bounds__(256) void spiral_wmma_kernel(
    const float* __restrict__ in,    // [N, C_IN]
    const int* __restrict__ idx,     // [N, S] (unused if !HAS_IDX)
    const float* __restrict__ w,     // [C_OUT, S*C_IN]
    const float* __restrict__ bias,  // [C_OUT]
    float* __restrict__ out,         // [N, C_OUT]
    int n_nodes) {
    static_assert(C_IN % 4 == 0, "K step of 4 must not cross a spiral step");
    static_assert(C_OUT % 16 == 0, "");
    constexpr int K = C_IN * S;

    const int lane = threadIdx.x & 31;
    const int wave = threadIdx.x >> 5;
    const int tile = blockIdx.x * 8 + wave;
    const int node_base = tile * 16;
    if (node_base >= n_nodes) return;  // wave-uniform: EXEC stays all-1s

    const int m  = lane & 15;  // node-in-tile (A rows) / out-channel (B cols)
    const int hi = lane >> 4;  // which K pair of the 4-wide step
    const int ob0 = blockIdx.y * (CHUNKS * 16);

    v8f acc[CHUNKS];
#pragma unroll
    for (int c = 0; c < CHUNKS; ++c)
#pragma unroll
        for (int r = 0; r < 8; ++r) acc[c][r] = 0.0f;

    // Per-chunk weight row base (B column n = lane&15), pre-offset by K pair.
    const float* wrow[CHUNKS];
#pragma unroll
    for (int c = 0; c < CHUNKS; ++c)
        wrow[c] = w + (size_t)(ob0 + c * 16 + m) * K + 2 * hi;

    const int* irow = HAS_IDX ? (idx + (size_t)(node_base + m) * S) : nullptr;

#pragma unroll 1
    for (int s = 0; s < S; ++s) {
        int row;
        if (HAS_IDX) row = irow[s];
        else         row = node_base + m;
        const float* arow = in + (size_t)row * C_IN + 2 * hi;
        const int kbase = s * C_IN;
#pragma unroll 4
        for (int c0 = 0; c0 < C_IN; c0 += 4) {
            v2f a = *(const v2f*)(arow + c0);  // gathered A fragment (b64)
            const int kg = kbase + c0;
#pragma unroll
            for (int c = 0; c < CHUNKS; ++c) {
                v2f bb = *(const v2f*)(wrow[c] + kg);  // B fragment (b64)
                acc[c] = __builtin_amdgcn_wmma_f32_16x16x4_f32(
                    false, a, false, bb, (short)0, acc[c], false, false);
            }
        }
    }

    // Epilogue: bias + ELU, coalesced 64B row stores per half-wave.
#pragma unroll
    for (int c = 0; c < CHUNKS; ++c) {
        const int ocol = ob0 + c * 16 + m;
        const float bv = bias[ocol];
#pragma unroll
        for (int r = 0; r < 8; ++r) {
            float v = acc[c][r] + bv;
            out[(size_t)(node_base + r + 8 * hi) * C_OUT + ocol] = elu(v);
        }
    }
}

// ---------------------------------------------------------------------------
// fc2 + log_softmax: [N,256] -> [N,12].  0.6 GFLOP; VALU, thread per node.
// Weights (12x256=12KB) are lane-uniform -> served from scalar/L1 caches.
// ---------------------------------------------------------------------------
__global__ __launch_bounds__(256) void fc2_logsoftmax_kernel(
    const float* __restrict__ h, const float* __restrict__ w,
    const float* __restrict__ b, float* __restrict__ out, int n) {
    int i = blockIdx.x * blockDim.x + threadIdx.x;
    if (i >= n) return;
    float acc[12];
#pragma unroll
    for (int o = 0; o < 12; ++o) acc[o] = b[o];
    const float* hr = h + (size_t)i * 256;
#pragma unroll 1
    for (int k = 0; k < 256; k += 4) {
        float4 hv = *(const float4*)(hr + k);
#pragma unroll
        for (int o = 0; o < 12; ++o) {
            float4 wv = *(const float4*)(w + o * 256 + k);
            acc[o] = fmaf(hv.x, wv.x, acc[o]);
            acc[o] = fmaf(hv.y, wv.y, acc[o]);
            acc[o] = fmaf(hv.z, wv.z, acc[o]);
            acc[o] = fmaf(hv.w, wv.w, acc[o]);
        }
    }
    float mx = acc[0];
#pragma unroll
    for (int o = 1; o < 12; ++o) mx = fmaxf(mx, acc[o]);
    float sum = 0.0f;
#pragma unroll
    for (int o = 0; o < 12; ++o) sum += __expf(acc[o] - mx);
    const float lse = mx + __logf(sum);
#pragma unroll
    for (int o = 0; o < 12; ++o) out[(size_t)i * 12 + o] = acc[o] - lse;
}

// ---------------------------------------------------------------------------
extern "C" void kernel_launch(void* const* d_in, const int* in_sizes, int n_in,
                              void* d_out, int out_size, void* d_ws,
                              size_t ws_size, hipStream_t stream) {
    const float* x     = (const float*)d_in[0];
    const int*   idx   = (const int*)d_in[1];
    const float* fc0_w = (const float*)d_in[2];
    const float* fc0_b = (const float*)d_in[3];
    const float* w1    = (const float*)d_in[4];
    const float* b1    = (const float*)d_in[5];
    const float* w2    = (const float*)d_in[6];
    const float* b2    = (const float*)d_in[7];
    const float* w3    = (const float*)d_in[8];
    const float* b3    = (const float*)d_in[9];
    const float* fc1_w = (const float*)d_in[10];
    const float* fc1_b = (const float*)d_in[11];
    const float* fc2_w = (const float*)d_in[12];
    const float* fc2_b = (const float*)d_in[13];
    float* out = (float*)d_out;

    const int N = in_sizes[0] / 3;  // 100000

    float* h0 = (float*)d_ws;            // [N,16]
    float* h1 = h0 + (size_t)N * 16;     // [N,32]
    float* h2 = h1 + (size_t)N * 32;     // [N,64]
    float* h3 = h2 + (size_t)N * 64;     // [N,128]
    float* h4 = h3 + (size_t)N * 128;    // [N,256]

    const dim3 blk(256);
    const int tiles = (N + 15) / 16;
    const int gx = (tiles + 7) / 8;  // 8 waves (tiles) per block

    fc0_elu_kernel<<<dim3((N + 255) / 256), blk, 0, stream>>>(x, fc0_w, fc0_b,
                                                              h0, N);
    spiral_wmma_kernel<16, 32, 16, true, 2>
        <<<dim3(gx, 1), blk, 0, stream>>>(h0, idx, w1, b1, h1, N);
    spiral_wmma_kernel<32, 64, 16, true, 4>
        <<<dim3(gx, 1), blk, 0, stream>>>(h1, idx, w2, b2, h2, N);
    spiral_wmma_kernel<64, 128, 16, true, 8>
        <<<dim3(gx, 1), blk, 0, stream>>>(h2, idx, w3, b3, h3, N);
    // fc1 as S=1 "conv": 256 outs = grid.y(4) x CHUNKS(4) x 16
    spiral_wmma_kernel<128, 256, 1, false, 4>
        <<<dim3(gx, 4), blk, 0, stream>>>(h3, nullptr, fc1_w, fc1_b, h4, N);
    fc2_logsoftmax_kernel<<<dim3((N + 255) / 256), blk, 0, stream>>>(
        h4, fc2_w, fc2_b, out, N);
}